// SoftDTW_45561013076219
// MI455X (gfx1250) — compile-verified
//
#include <hip/hip_runtime.h>

// Problem constants (match reference)
#define BB   32
#define NN   512     // N == M == 512
#define DIMK 64
#define CELLS_PER_B (NN * NN)        // 262144
#define ROWS_TOTAL  (2 * BB * NN)    // 32768 rows of length 64 (X then Y)

typedef float v2f __attribute__((ext_vector_type(2)));
typedef float v8f __attribute__((ext_vector_type(8)));

// Offset of anti-diagonal s (s = i + j, 0-based, s in [0, 1022]) in
// diagonal-major storage of a 512x512 matrix.
__device__ __forceinline__ int diag_off(int s) {
    // lengths: 1..512 for s=0..511, then 511..1
    return (s <= 511) ? ((s * (s + 1)) >> 1)
                      : (131328 + (((1535 - s) * (s - 512)) >> 1));
}

// ---------------------------------------------------------------------------
// Kernel 1: row squared-norms. One wave32 per row, lane loads float2,
// cross-lane reduce, lane 0 writes. Rows 0..16383 -> x2, 16384..32767 -> y2.
// ---------------------------------------------------------------------------
__global__ __launch_bounds__(256) void sqnorm_kernel(
    const float* __restrict__ X, const float* __restrict__ Y,
    float* __restrict__ x2, float* __restrict__ y2) {
    int wave = (blockIdx.x * 256 + threadIdx.x) >> 5;
    int lane = threadIdx.x & 31;
    if (wave >= ROWS_TOTAL) return;

    const float* src;
    float* dst;
    if (wave < BB * NN) {
        src = X + (size_t)wave * DIMK;
        dst = x2 + wave;
    } else {
        int r = wave - BB * NN;
        src = Y + (size_t)r * DIMK;
        dst = y2 + r;
    }
    v2f v = *(const v2f*)(src + lane * 2);
    float s = v.x * v.x + v.y * v.y;
#pragma unroll
    for (int m = 16; m > 0; m >>= 1) s += __shfl_xor(s, m, 32);
    if (lane == 0) *dst = s;
}

// ---------------------------------------------------------------------------
// Kernel 2: D = x2 + y2 - 2*X@Y^T via V_WMMA_F32_16X16X4_F32.
// One wave per 16x16 output tile; 16 k-steps of K=4.
// Output scattered into diagonal-major layout (coalesced reads for the scan).
//
// WMMA f32 layouts (wave32, ISA 7.12.2):
//   A 16x4:  lane<16: M=lane, v0=K0,v1=K1 ; lane>=16: M=lane-16, v0=K2,v1=K3
//   B 4x16:  lane<16: N=lane, v0=K0,v1=K1 ; lane>=16: N=lane-16, v0=K2,v1=K3
//   C 16x16: vgpr r:  lane<16: (M=r,  N=lane) ; lane>=16: (M=r+8, N=lane-16)
// ---------------------------------------------------------------------------
__global__ __launch_bounds__(256) void pairdist_wmma_kernel(
    const float* __restrict__ X, const float* __restrict__ Y,
    const float* __restrict__ x2, const float* __restrict__ y2,
    float* __restrict__ Ddiag) {
    int wave = (blockIdx.x * 256 + threadIdx.x) >> 5;   // [0, 32768)
    int lane = threadIdx.x & 31;
    int b  = wave >> 10;
    int ti = (wave >> 5) & 31;   // row tile
    int tj = wave & 31;          // col tile

    const float* Xb = X + ((size_t)b * NN + ti * 16) * DIMK;
    const float* Yb = Y + ((size_t)b * NN + tj * 16) * DIMK;

    int mn = lane & 15;
    int kh = (lane >> 4) * 2;    // 0 or 2

    v8f acc = {};
#pragma unroll
    for (int kk = 0; kk < 16; ++kk) {
        int kb = kk * 4 + kh;
        v2f a  = *(const v2f*)(Xb + mn * DIMK + kb);
        v2f bv = *(const v2f*)(Yb + mn * DIMK + kb);
        acc = __builtin_amdgcn_wmma_f32_16x16x4_f32(
            false, a, false, bv, (short)0, acc, false, false);
    }

    int hi8 = (lane >> 4) << 3;          // 0 or 8
    int n   = tj * 16 + (lane & 15);
    float yn = y2[(size_t)b * NN + n];
    float* Db = Ddiag + (size_t)b * CELLS_PER_B;
#pragma unroll
    for (int r = 0; r < 8; ++r) {
        int m = ti * 16 + r + hi8;
        float d = x2[(size_t)b * NN + m] + yn - 2.0f * acc[r];
        int s = m + n;
        int p = m - ((s > 511) ? (s - 511) : 0);
        Db[diag_off(s) + p] = d;
    }
}

// ---------------------------------------------------------------------------
// Kernel 3: soft-DTW wavefront scan (gamma = 1). One 512-thread WG per batch.
// Three rotating diagonal buffers in LDS, indexed by row i1 (1-based), with
// +inf borders maintained exactly as in the padded-R reference.
// ---------------------------------------------------------------------------
__global__ __launch_bounds__(512) void softdtw_scan_kernel(
    const float* __restrict__ Ddiag, float* __restrict__ out) {
    __shared__ float bufs[3 * 513];
    const float INF = __builtin_inff();
    int b   = blockIdx.x;
    int tid = threadIdx.x;

    for (int i = tid; i < 3 * 513; i += 512) bufs[i] = INF;
    __syncthreads();
    if (tid == 0) bufs[0] = 0.0f;       // R[0,0] = 0 (diag-0 buffer)
    __syncthreads();

    float* Rm2 = bufs;                   // diag t-2
    float* Rm1 = bufs + 513;             // diag t-1
    float* Rc  = bufs + 2 * 513;         // diag t (being written)

    const float* Db = Ddiag + (size_t)b * CELLS_PER_B;

    for (int t = 2; t <= 2 * NN; ++t) {
        int lo  = (t - NN > 1) ? (t - NN) : 1;
        int hi  = (t - 1 < NN) ? (t - 1) : NN;
        int len = hi - lo + 1;
        if (tid < len) {
            int i1 = lo + tid;
            float ra = Rm2[i1 - 1];      // R[i-1, j-1]
            float rb = Rm1[i1 - 1];      // R[i-1, j]
            float rc = Rm1[i1];          // R[i,   j-1]
            float r0 = -ra, r1 = -rb, r2 = -rc;
            float rmax = fmaxf(fmaxf(r0, r1), r2);
            float rsum = __expf(r0 - rmax) + __expf(r1 - rmax) + __expf(r2 - rmax);
            float softmin = -(__logf(rsum) + rmax);
            float d = Db[diag_off(t - 2) + tid];
            Rc[i1] = d + softmin;
        }
        // maintain +inf borders: cell (0, t) at index 0, cell (t, 0) at index t
        if (t <= NN) {
            if (tid == 0)  Rc[0] = INF;
            if (tid == 31) Rc[t] = INF;
        }
        __syncthreads();
        float* tmp = Rm2; Rm2 = Rm1; Rm1 = Rc; Rc = tmp;
    }
    if (tid == 0) out[b] = Rm1[NN];      // R[512, 512] on diag 1024
}

// ---------------------------------------------------------------------------
extern "C" void kernel_launch(void* const* d_in, const int* in_sizes, int n_in,
                              void* d_out, int out_size, void* d_ws, size_t ws_size,
                              hipStream_t stream) {
    const float* X = (const float*)d_in[0];   // (32, 512, 64) f32
    const float* Y = (const float*)d_in[1];   // (32, 512, 64) f32
    float* out = (float*)d_out;               // (32,) f32

    char*  ws    = (char*)d_ws;
    float* Ddiag = (float*)ws;                                       // 33.55 MB
    float* x2    = (float*)(ws + (size_t)BB * CELLS_PER_B * 4);      // 64 KB
    float* y2    = x2 + BB * NN;                                     // 64 KB

    // 1) row squared norms: 32768 rows, 8 waves/block
    sqnorm_kernel<<<ROWS_TOTAL / 8, 256, 0, stream>>>(X, Y, x2, y2);

    // 2) WMMA pairwise distances into diagonal-major D: 32768 tiles, 8/block
    pairdist_wmma_kernel<<<(BB * 32 * 32) / 8, 256, 0, stream>>>(X, Y, x2, y2, Ddiag);

    // 3) anti-diagonal soft-DTW scan: one workgroup per batch
    softdtw_scan_kernel<<<BB, 512, 0, stream>>>(Ddiag, out);
}